// top_LSTM_60189671686547
// MI455X (gfx1250) — compile-verified
//
#include <hip/hip_runtime.h>
#include <cmath>

// ---------------------------------------------------------------------------
// LSTM (B=64, T=512, D=H=256, O=256) for MI455X / gfx1250, wave32.
//
// Kernel 1 (parallel):  P = X @ W_i + bias  (32768 x 1024 x 256 fp32 GEMM via
//                       V_WMMA_F32_16X16X4_F32), staged into the i/f/g/o slots
//                       of states_seq inside d_out (later overwritten by the
//                       activated gates at the same position -> no workspace).
// Kernel 2 (persistent, 64 WGs x 128 thr): 512 sequential steps. Each WG owns
//   one (16-batch x 16-hidden) tile; its 4 waves own gates i,f,g,o. W_h
//   fragments live in registers for all 512 steps (64 x float2 = 128 VGPRs).
//   Per step: C := P-tile (global), C += h_{t-1} @ W_h (64 WMMAs), activate,
//   exchange gates through LDS, update c/h, write 6 state planes, device-wide
//   barrier (atomic counter in d_ws, re-zeroed every launch by init kernel).
//   Epilogue: out = h_T @ fc_w^T + fc_b (one WMMA tile per WG, wave 0).
// ---------------------------------------------------------------------------

typedef float v2f __attribute__((ext_vector_type(2)));
typedef float v8f __attribute__((ext_vector_type(8)));

#define T_SEQ 512
#define DIM   256
#define HID   256
#define G4    1024   // 4*HID
#define NWG   64     // persistent workgroups (4 batch blocks x 16 hidden blocks)

// states_seq layout: (B, 6, T, H); s: 0=h 1=c 2=i 3=f 4=g 5=o
__device__ __forceinline__ size_t sidx(int b, int s, int t, int n) {
  return (((size_t)b * 6 + s) * T_SEQ + t) * HID + n;
}

__global__ void lstm_init_barrier(unsigned* __restrict__ bar) {
  if (threadIdx.x == 0) *bar = 0u;
}

// ---------------- Kernel 1: P = X @ W_i + bias -----------------------------
__global__ __launch_bounds__(128) void lstm_preact_kernel(
    const float* __restrict__ x, const float* __restrict__ W_i,
    const float* __restrict__ bias, float* __restrict__ states) {
  const int lane = threadIdx.x & 31;
  const int wave = threadIdx.x >> 5;
  const int tile = blockIdx.x * 4 + wave;   // 4 waves share one row tile (A reuse)
  const int rowTile = tile >> 6;            // 0..2047  (B*T/16)
  const int colTile = tile & 63;            // 0..63    (4H/16)
  const int l15  = lane & 15;
  const int half = lane >> 4;               // ISA: lanes 16..31 carry K+2
  const int m0   = rowTile * 16;
  const int col  = colTile * 16 + l15;      // 0..1023

  const float* arow = x + (size_t)(m0 + l15) * DIM + 2 * half;
  const float* bcol = W_i + (size_t)(2 * half) * G4 + col;

  v8f acc = {};
#pragma unroll 8
  for (int kc = 0; kc < 64; ++kc) {         // K = 256 in chunks of 4
    v2f a = *(const v2f*)(arow + 4 * kc);   // A 16x4: {K, K+1} at this lane
    v2f b;                                  // B 4x16: rows striped over lanes
    b.x = bcol[(size_t)(4 * kc) * G4];
    b.y = bcol[(size_t)(4 * kc + 1) * G4];
    acc = __builtin_amdgcn_wmma_f32_16x16x4_f32(false, a, false, b,
                                                (short)0, acc, false, false);
  }
  const float bv   = bias[col];
  const int   gate = col >> 8;              // which of i/f/g/o
  const int   n    = col & 255;
#pragma unroll
  for (int v = 0; v < 8; ++v) {             // C/D: VGPR v -> row v + 8*half
    int r  = m0 + v + 8 * half;             // flat (b*T + t)
    int b_ = r >> 9;
    int t  = r & (T_SEQ - 1);
    states[sidx(b_, 2 + gate, t, n)] = acc[v] + bv;
  }
}

// ---------------- Kernel 2: persistent recurrence --------------------------
__global__ __launch_bounds__(128) void lstm_seq_kernel(
    const float* __restrict__ W_h, const float* __restrict__ fc_w,
    const float* __restrict__ fc_b, float* __restrict__ out,
    float* __restrict__ states, unsigned* __restrict__ barrier) {
  __shared__ float act[4][16][17];          // activated gate tiles (padded)
  __shared__ float cbuf[16][16];            // persistent cell state tile

  const int lane = threadIdx.x & 31;
  const int g    = threadIdx.x >> 5;        // wave -> gate
  const int l15  = lane & 15;
  const int half = lane >> 4;
  const int rb   = blockIdx.x >> 4;         // batch block 0..3
  const int cb   = blockIdx.x & 15;         // hidden block 0..15

  const int n     = cb * 16 + l15;          // hidden column
  const int whcol = g * HID + n;            // column in W_h (H x 4H)
  const int arow  = rb * 16 + l15;          // batch row for A fragments

  // W_h B-fragments cached in registers for all 512 steps (128 VGPRs).
  v2f bfrag[64];
#pragma unroll
  for (int kc = 0; kc < 64; ++kc) {
    int k = 4 * kc + 2 * half;
    bfrag[kc].x = W_h[(size_t)k * G4 + whcol];
    bfrag[kc].y = W_h[(size_t)(k + 1) * G4 + whcol];
  }

  // c_0 = 0
  {
    int e = threadIdx.x;
    cbuf[e >> 4][e & 15] = 0.0f;
    e += 128;
    cbuf[e >> 4][e & 15] = 0.0f;
  }
  __syncthreads();

  for (int t = 0; t < T_SEQ; ++t) {
    // C := preactivation tile (x@W_i + bias, staged by kernel 1)
    v8f acc;
#pragma unroll
    for (int v = 0; v < 8; ++v)
      acc[v] = states[sidx(rb * 16 + v + 8 * half, 2 + g, t, n)];

    if (t + 1 < T_SEQ)  // warm L2->L0 for next step's P tile
      __builtin_prefetch(&states[sidx(rb * 16 + 8 * half, 2 + g, t + 1, n)], 0, 1);

    if (t > 0) {        // C += h_{t-1} @ W_h  (h read from states plane 0)
      const float* hrow = states + sidx(arow, 0, t - 1, 2 * half);
#pragma unroll
      for (int kc = 0; kc < 64; ++kc) {
        v2f a = *(const v2f*)(hrow + 4 * kc);
        acc = __builtin_amdgcn_wmma_f32_16x16x4_f32(false, a, false, bfrag[kc],
                                                    (short)0, acc, false, false);
      }
    }

    // Activate this wave's gate; publish to LDS + global (overwrites P slot).
#pragma unroll
    for (int v = 0; v < 8; ++v) {
      float pre = acc[v];
      float a = (g == 2) ? tanhf(pre) : 1.0f / (1.0f + __expf(-pre));
      int row = v + 8 * half;
      act[g][row][l15] = a;
      states[sidx(rb * 16 + row, 2 + g, t, n)] = a;
    }
    __syncthreads();

    // Elementwise cell/hidden update: 256 elems / 128 threads.
    for (int e = threadIdx.x; e < 256; e += 128) {
      int row = e >> 4, col = e & 15;
      float iv = act[0][row][col];
      float fv = act[1][row][col];
      float gv = act[2][row][col];
      float ov = act[3][row][col];
      float cn = fv * cbuf[row][col] + iv * gv;
      float hv = ov * tanhf(cn);
      cbuf[row][col] = cn;
      int b_ = rb * 16 + row;
      int nn = cb * 16 + col;
      states[sidx(b_, 0, t, nn)] = hv;
      states[sidx(b_, 1, t, nn)] = cn;
    }

    // Device-wide barrier: all 64 WGs must see h_t before step t+1.
    __syncthreads();
    if (threadIdx.x == 0) {
      __threadfence();                      // release h_t
      atomicAdd(barrier, 1u);
      const unsigned target = (unsigned)NWG * (unsigned)(t + 1);
      while (__hip_atomic_load(barrier, __ATOMIC_ACQUIRE,
                               __HIP_MEMORY_SCOPE_AGENT) < target)
        __builtin_amdgcn_s_sleep(1);
    }
    __syncthreads();
    __threadfence();                        // acquire side for all threads
  }

  // Epilogue: out = h_{T-1} @ fc_w^T + fc_b. One 16x16 tile per WG (wave 0;
  // g is wave-uniform so EXEC stays all-ones inside the branch).
  if (g == 0) {
    v8f acc = {};
    const float* hrow = states + sidx(arow, 0, T_SEQ - 1, 2 * half);
    const int o = cb * 16 + l15;
#pragma unroll 8
    for (int kc = 0; kc < 64; ++kc) {
      v2f a = *(const v2f*)(hrow + 4 * kc);
      v2f b = *(const v2f*)(fc_w + (size_t)o * HID + 4 * kc + 2 * half); // B[k][o]=fc_w[o][k]
      acc = __builtin_amdgcn_wmma_f32_16x16x4_f32(false, a, false, b,
                                                  (short)0, acc, false, false);
    }
    const float bv = fc_b[o];
#pragma unroll
    for (int v = 0; v < 8; ++v)
      out[(size_t)(rb * 16 + v + 8 * half) * 256 + o] = acc[v] + bv;
  }
}

// ---------------------------------------------------------------------------
extern "C" void kernel_launch(void* const* d_in, const int* in_sizes, int n_in,
                              void* d_out, int out_size, void* d_ws, size_t ws_size,
                              hipStream_t stream) {
  const float* x    = (const float*)d_in[0];
  const float* W_i  = (const float*)d_in[1];
  const float* W_h  = (const float*)d_in[2];
  const float* bias = (const float*)d_in[3];
  const float* fc_w = (const float*)d_in[4];
  const float* fc_b = (const float*)d_in[5];

  float* out    = (float*)d_out;            // (64, 256)
  float* states = out + 64 * 256;           // (64, 6, 512, 256)
  unsigned* bar = (unsigned*)d_ws;          // 4-byte grid barrier counter

  lstm_init_barrier<<<1, 32, 0, stream>>>(bar);
  // 2048 row tiles x 64 col tiles, 4 waves (tiles) per 128-thread block.
  lstm_preact_kernel<<<32768, 128, 0, stream>>>(x, W_i, bias, states);
  // 64 persistent WGs (trivially co-resident on MI455X) run the recurrence.
  lstm_seq_kernel<<<NWG, 128, 0, stream>>>(W_h, fc_w, fc_b, out, states, bar);
}